// SCVC_22986664968208
// MI455X (gfx1250) — compile-verified
//
#include <hip/hip_runtime.h>

// SCVC on gfx1250: each 16x16 (p,q) output tile for fixed (n,g,o) is computed
// by a single V_WMMA_F32_16X16X4_F32 (K=4 rank-3 trick):
//   A[p,:] = { C*x[p],  a[p],  1, 0 }   (16x4)
//   B[:,q] = { y[q],    1,  b[q], 0 }   (4x16)
//   D[p,q] = C*x[p]*y[q] + a[p] + b[q]  == reference, full f32 precision.
// Kernel is store-bandwidth bound (268 MB out, ~1 FLOP/byte, ~11.5us floor
// at 23.3 TB/s). Output exceeds the 192 MB L2 and is never re-read, so
// stores are non-temporal (TH=NT) to avoid flushing the L2-resident x/y rows.

typedef __attribute__((ext_vector_type(2))) float v2f;
typedef __attribute__((ext_vector_type(8))) float v8f;

#define N_   8
#define G_   8
#define OG_  16
#define DX_  256
#define DY_  256
#define TP_  (DX_ / 16)
#define TQ_  (DY_ / 16)

__global__ __launch_bounds__(256) void scvc_wmma_kernel(
    const float* __restrict__ x,   // (N, G, DX)
    const float* __restrict__ y,   // (N, G, DY)
    const float* __restrict__ C,   // (G, OG)
    const float* __restrict__ wA,  // (G, OG, 2)
    const float* __restrict__ wB,  // (G, OG, 2)
    float* __restrict__ out)       // (N, G*OG, DX, DY)
{
  const int wid  = blockIdx.x * 8 + (threadIdx.x >> 5);
  const int lane = threadIdx.x & 31;
  const int half = lane >> 4;   // 0: lanes 0-15 (K=0/1 rows), 1: lanes 16-31 (K=2/3 rows)
  const int l    = lane & 15;

  // wid -> (n, g, o, p-tile, q-tile)
  int t = wid;
  const int qt = t % TQ_; t /= TQ_;
  const int pt = t % TP_; t /= TP_;
  const int o  = t % OG_; t /= OG_;
  const int g  = t % G_;  t /= G_;
  const int n  = t;

  const int p0 = pt * 16, q0 = qt * 16;
  const int go = g * OG_ + o;

  // wave-uniform scalars
  const float Cgo = C[go];
  const float wA0 = wA[go * 2 + 0], wA1 = wA[go * 2 + 1];
  const float wB0 = wB[go * 2 + 0], wB1 = wB[go * 2 + 1];

  const float* __restrict__ xb = x + (size_t)(n * G_ + g) * DX_;
  const float* __restrict__ yb = y + (size_t)(n * G_ + g) * DY_;

  const int p = p0 + l;
  const int q = q0 + l;
  const float xv  = xb[p];
  const float xlo = (p > 0)       ? xb[p - 1] : 0.0f;   // x[p-1], zero pad
  const float xhi = (p < DX_ - 1) ? xb[p + 1] : 0.0f;   // x[p+1], zero pad
  const float yv  = yb[q];
  const float ylo = (q > 0)       ? yb[q - 1] : 0.0f;
  const float yhi = (q < DY_ - 1) ? yb[q + 1] : 0.0f;

  const float aP = wA0 * xlo + wA1 * xhi;  // a[n,g,o,p]
  const float bQ = wB0 * ylo + wB1 * yhi;  // b[n,g,o,q]

  // A 16x4 f32 layout: vgpr0 = K=0 (lanes 0-15) / K=2 (lanes 16-31)
  //                    vgpr1 = K=1 (lanes 0-15) / K=3 (lanes 16-31)
  v2f A, B;
  A.x = half ? 1.0f : (Cgo * xv);  // K=0: C*x[p]   | K=2: 1
  A.y = half ? 0.0f : aP;          // K=1: a[p]     | K=3: 0
  // B 4x16 f32 layout: vgpr0 = row K=0 (lanes 0-15) / row K=2 (lanes 16-31)
  //                    vgpr1 = row K=1 (lanes 0-15) / row K=3 (lanes 16-31)
  B.x = half ? bQ : yv;            // K=0: y[q]     | K=2: b[q]
  B.y = half ? 0.0f : 1.0f;        // K=1: 1        | K=3: 0

  v8f acc = {};
  // (neg_a, A, neg_b, B, c_mod, C, reuse_a, reuse_b)
  acc = __builtin_amdgcn_wmma_f32_16x16x4_f32(
      false, A, false, B, (short)0, acc, false, false);

  // D layout: VGPR r -> row (r + 8*half), col = l
  float* __restrict__ ob =
      out + ((size_t)(n * (G_ * OG_) + go) * DX_ + p0) * DY_ + q0;
#pragma unroll
  for (int r = 0; r < 8; ++r) {
    // streaming output, never re-read: non-temporal store (th:TH_STORE_NT)
    __builtin_nontemporal_store(acc[r], &ob[(size_t)(r + half * 8) * DY_ + l]);
  }
}

extern "C" void kernel_launch(void* const* d_in, const int* in_sizes, int n_in,
                              void* d_out, int out_size, void* d_ws, size_t ws_size,
                              hipStream_t stream) {
  const float* x  = (const float*)d_in[0];
  const float* y  = (const float*)d_in[1];
  const float* C  = (const float*)d_in[2];
  const float* wA = (const float*)d_in[3];
  const float* wB = (const float*)d_in[4];
  float* out = (float*)d_out;

  // one wave per 16x16 tile: N*G*OG*TP*TQ = 8*8*16*16*16 = 262144 waves
  const int total_waves = N_ * G_ * OG_ * TP_ * TQ_;
  const int waves_per_block = 8;               // 256 threads, wave32
  const int blocks = total_waves / waves_per_block;  // 32768, exact cover

  scvc_wmma_kernel<<<blocks, 256, 0, stream>>>(x, y, C, wA, wB, out);
}